// InterSampleAttention_36661840839198
// MI455X (gfx1250) — compile-verified
//
#include <hip/hip_runtime.h>
#include <hip/hip_bf16.h>
#include <math.h>

#define BLK   256
#define L_SEQ 16
#define DM    256
#define DI    512
#define DS    16
#define DR    16
#define DBCW  (DR + 2*DS)   // 48
#define NSP   1024          // 32*32 spatial positions

typedef __attribute__((ext_vector_type(2))) float v2f;
typedef __attribute__((ext_vector_type(8))) float v8f;

// One 16x16 output tile of A(16xK) @ B(KxN-tile), fp32 WMMA, K%4==0.
// A: LDS row-major (lda even), B: global row-major (ldb), tile at column n0.
// ISA layouts: A 16x4 -> lane m=lane&15, K = 2*(lane>>4)+vgpr (adjacent pair);
//              B 4x16 -> lane n=lane&15, row K = 2*(lane>>4)+vgpr;
//              C/D     -> vgpr i holds rows i and i+8 (lane halves), col = lane&15.
__device__ __forceinline__ v8f wmma_gemm_tile(const float* __restrict__ A, int lda,
                                              const float* __restrict__ B, int ldb,
                                              int K, int n0, int lane) {
  v8f acc = {0.f, 0.f, 0.f, 0.f, 0.f, 0.f, 0.f, 0.f};
  const int half = lane >> 4;   // 0 or 1
  const int mr   = lane & 15;   // A row / B column within tile
  for (int k = 0; k < K; k += 4) {
    const int ka = k + 2 * half;
    v2f a = *(const v2f*)(A + mr * lda + ka);   // K=ka, ka+1 (8B aligned: lda,ka even)
    v2f b;
    b.x = B[(size_t)(ka    ) * ldb + n0 + mr];
    b.y = B[(size_t)(ka + 1) * ldb + n0 + mr];
    acc = __builtin_amdgcn_wmma_f32_16x16x4_f32(
        /*neg_a=*/false, a, /*neg_b=*/false, b,
        /*c_mod=*/(short)0, acc, /*reuse_a=*/false, /*reuse_b=*/false);
  }
  return acc;
}

__device__ __forceinline__ void store_tile_lds(float* C, int ldc, int n0, int lane, v8f acc) {
  const int half = lane >> 4;
  const int n    = lane & 15;
#pragma unroll
  for (int i = 0; i < 8; ++i)
    C[(i + 8 * half) * ldc + n0 + n] = acc[i];
}

__device__ __forceinline__ float silu_f(float v) { return v / (1.f + expf(-v)); }

__global__ void __launch_bounds__(BLK)
mamba_residual_kernel(const float* __restrict__ feature, const float* __restrict__ norm_w,
                      const float* __restrict__ w_in,    const float* __restrict__ conv_w,
                      const float* __restrict__ conv_b,  const float* __restrict__ w_xproj,
                      const float* __restrict__ w_dt,    const float* __restrict__ b_dt,
                      const float* __restrict__ A_log,   const float* __restrict__ Dp,
                      const float* __restrict__ w_out,   float* __restrict__ out) {
  extern __shared__ float smem[];
  float* sX   = smem;                 // 16*256  original x (residual)
  float* sXN  = sX   + L_SEQ * DM;    // 16*256  rmsnorm(x)
  float* sXM  = sXN  + L_SEQ * DM;    // 16*512  xm (raw -> conv+silu in place)
  float* sRES = sXM  + L_SEQ * DI;    // 16*512  gate branch
  float* sDel = sRES + L_SEQ * DI;    // 16*512  softplus delta
  float* sY   = sDel + L_SEQ * DI;    // 16*512  scan output (gated)
  float* sDBC = sY   + L_SEQ * DI;    // 16*48   [delta_lowrank | B | C]

  const int b    = blockIdx.x;        // spatial position 0..1023
  const int tid  = threadIdx.x;
  const int wave = tid >> 5;
  const int lane = tid & 31;

  // ---- Phase A: gather x[b] (strided over channel axis) + RMSNorm ----
  for (int i = tid; i < L_SEQ * DM; i += BLK) {
    const int l = i >> 8, c = i & (DM - 1);
    sX[i] = feature[(size_t)l * DM * NSP + (size_t)c * NSP + b];
  }
  __syncthreads();
  for (int l = wave; l < L_SEQ; l += 8) {
    float s = 0.f;
    for (int c = lane; c < DM; c += 32) { const float v = sX[l * DM + c]; s += v * v; }
#pragma unroll
    for (int off = 16; off > 0; off >>= 1) s += __shfl_xor(s, off, 32);
    const float rstd = rsqrtf(s * (1.f / DM) + 1e-5f);
    for (int c = lane; c < DM; c += 32)
      sXN[l * DM + c] = sX[l * DM + c] * rstd * norm_w[c];
  }
  __syncthreads();

  // ---- Phase B: xr = xn @ w_in  (16x256 @ 256x1024) -> xm | res ----
  for (int t = wave * 8; t < wave * 8 + 8; ++t) {
    const int n0 = t * 16;
    v8f acc = wmma_gemm_tile(sXN, DM, w_in, 2 * DI, DM, n0, lane);
    if (n0 < DI) store_tile_lds(sXM, DI, n0, lane, acc);
    else         store_tile_lds(sRES, DI, n0 - DI, lane, acc);
  }
  __syncthreads();

  // ---- Phase C: depthwise causal conv (k=3, pad left 2) + bias + silu ----
  for (int d = tid; d < DI; d += BLK) {
    float r[L_SEQ];
#pragma unroll
    for (int l = 0; l < L_SEQ; ++l) r[l] = sXM[l * DI + d];
    const float w0 = conv_w[d * 3 + 0], w1 = conv_w[d * 3 + 1], w2 = conv_w[d * 3 + 2];
    const float cb = conv_b[d];
#pragma unroll
    for (int l = 0; l < L_SEQ; ++l) {
      float v = w2 * r[l] + cb;
      if (l >= 1) v += w1 * r[l - 1];
      if (l >= 2) v += w0 * r[l - 2];
      sXM[l * DI + d] = silu_f(v);
    }
  }
  __syncthreads();

  // ---- Phase D: dbc = xm @ w_xproj (16x512 @ 512x48) ----
  if (wave < 3) {
    const int n0 = wave * 16;
    v8f acc = wmma_gemm_tile(sXM, DI, w_xproj, DBCW, DI, n0, lane);
    store_tile_lds(sDBC, DBCW, n0, lane, acc);
  }
  __syncthreads();

  // ---- Phase E: delta = softplus(dbc[:, :16] @ w_dt + b_dt) (16x16 @ 16x512) ----
  for (int t = wave * 4; t < wave * 4 + 4; ++t) {
    const int n0 = t * 16;
    v8f acc = wmma_gemm_tile(sDBC, DBCW, w_dt, DI, DR, n0, lane);
    const int half = lane >> 4, n = lane & 15;
#pragma unroll
    for (int i = 0; i < 8; ++i) {
      const int M = i + 8 * half, c = n0 + n;
      float v = acc[i] + b_dt[c];
      v = (v > 20.f) ? v : log1pf(expf(v));
      sDel[M * DI + c] = v;
    }
  }
  __syncthreads();

  // ---- Phase F: selective scan over L (h-state in registers) + gate ----
  for (int d = tid; d < DI; d += BLK) {
    float Ar[DS], h[DS];
#pragma unroll
    for (int n = 0; n < DS; ++n) { Ar[n] = -expf(A_log[d * DS + n]); h[n] = 0.f; }
    const float Dd = Dp[d];
    for (int l = 0; l < L_SEQ; ++l) {
      const float dl = sDel[l * DI + d];
      const float xm = sXM[l * DI + d];
      const float du = dl * xm;
      float y = 0.f;
#pragma unroll
      for (int n = 0; n < DS; ++n) {
        const float dA = expf(dl * Ar[n]);
        h[n] = dA * h[n] + du * sDBC[l * DBCW + DR + n];
        y += h[n] * sDBC[l * DBCW + DR + DS + n];
      }
      const float res = sRES[l * DI + d];
      sY[l * DI + d] = (y + xm * Dd) * silu_f(res);
    }
  }
  __syncthreads();

  // ---- Phase H: out = y @ w_out + x  (16x512 @ 512x256), scatter to NCHW ----
  for (int t = wave * 2; t < wave * 2 + 2; ++t) {
    const int n0 = t * 16;
    v8f acc = wmma_gemm_tile(sY, DI, w_out, DM, DI, n0, lane);
    const int half = lane >> 4, n = lane & 15;
#pragma unroll
    for (int i = 0; i < 8; ++i) {
      const int M = i + 8 * half, c = n0 + n;
      out[(size_t)M * DM * NSP + (size_t)c * NSP + b] = acc[i] + sX[M * DM + c];
    }
  }
}

extern "C" void kernel_launch(void* const* d_in, const int* in_sizes, int n_in,
                              void* d_out, int out_size, void* d_ws, size_t ws_size,
                              hipStream_t stream) {
  (void)in_sizes; (void)n_in; (void)out_size; (void)d_ws; (void)ws_size;
  const float* feature = (const float*)d_in[0];
  const float* norm_w  = (const float*)d_in[1];
  const float* w_in    = (const float*)d_in[2];
  const float* conv_w  = (const float*)d_in[3];
  const float* conv_b  = (const float*)d_in[4];
  const float* w_xproj = (const float*)d_in[5];
  const float* w_dt    = (const float*)d_in[6];
  const float* b_dt    = (const float*)d_in[7];
  const float* A_log   = (const float*)d_in[8];
  const float* Dp      = (const float*)d_in[9];
  const float* w_out   = (const float*)d_in[10];
  float* out = (float*)d_out;

  const size_t shmem = (size_t)(2 * L_SEQ * DM + 4 * L_SEQ * DI + L_SEQ * DBCW) * sizeof(float);
  // > 64 KB dynamic LDS: opt in (idempotent; safe under graph capture).
  hipFuncSetAttribute((const void*)mamba_residual_kernel,
                      hipFuncAttributeMaxDynamicSharedMemorySize, (int)shmem);

  mamba_residual_kernel<<<dim3(NSP), dim3(BLK), shmem, stream>>>(
      feature, norm_w, w_in, conv_w, conv_b, w_xproj, w_dt, b_dt, A_log, Dp, w_out, out);
}